// OTAD_35639638622408
// MI455X (gfx1250) — compile-verified
//
#include <hip/hip_runtime.h>
#include <cfloat>

typedef float v2f __attribute__((ext_vector_type(2)));
typedef float v8f __attribute__((ext_vector_type(8)));

#define TOPK   10
#define D_DIM  384
#define BK     32      // k-chunk staged in LDS
#define BN     128     // columns (data rows) per block
#define NROWS  128     // query rows (B) — all handled by every block
#define AST    36      // LDS stride for staging tiles (even, 16B-friendly, conflict-free)
#define SST    129     // LDS stride for score tile (odd -> conflict-free row scans)

// Branchless sorted insertion into an ascending top-K list held in registers.
// Static indices only -> compiler keeps v[]/ix[] in VGPRs.
__device__ __forceinline__ void topk_insert(float (&v)[TOPK], int (&ix)[TOPK],
                                            float s, int gi) {
    if (s < v[TOPK - 1]) {
#pragma unroll
        for (int q = TOPK - 1; q >= 1; --q) {
            bool below_prev = s < v[q - 1];
            float nv = below_prev ? v[q - 1] : s;
            int   ni = below_prev ? ix[q - 1] : gi;
            bool take = s < v[q];
            v[q]  = take ? nv : v[q];
            ix[q] = take ? ni : ix[q];
        }
        if (s < v[0]) { v[0] = s; ix[0] = gi; }
    }
}

// Kernel 1: fused f32-WMMA GEMM (targets x data^T) + ||d||^2 + per-block top-K.
// Each block: all 128 query rows x BN=128 data columns. 8 waves, wave w owns
// query-row tile w (16 rows) across all 8 column tiles.
__global__ __launch_bounds__(256)
void OTAD_dist_topk(const float* __restrict__ targets,  // [128, 384]
                    const float* __restrict__ data,     // [N, 384]
                    float* __restrict__ cand_val,       // [128, nb, 10]
                    int*   __restrict__ cand_idx,       // [128, nb, 10]
                    int N, int nb)
{
    extern __shared__ float smem[];
    float* As  = smem;               // [NROWS][AST]  staging: targets chunk
    float* Bs  = smem + NROWS * AST; // [BN][AST]     staging: data chunk
    float* Ss  = smem;               // [NROWS][SST]  score tile (reuses staging)
    float* d2s = smem + NROWS * SST; // [BN]          per-column ||d||^2

    const int tid  = threadIdx.x;
    const int wave = tid >> 5;
    const int lane = tid & 31;
    const int half = lane >> 4;   // K-pair select for 16x16x4 f32 fragments
    const int lm   = lane & 15;   // M (A) / N (B,C) within tile
    const int col0 = blockIdx.x * BN;

    v8f acc[8];
    {
        v8f z = {0.f, 0.f, 0.f, 0.f, 0.f, 0.f, 0.f, 0.f};
#pragma unroll
        for (int i = 0; i < 8; ++i) acc[i] = z;
    }

    const int srow  = tid >> 1;          // 0..127: staging row
    const int skoff = (tid & 1) << 4;    // 0 or 16: staging k-offset
    const float4 fzero = make_float4(0.f, 0.f, 0.f, 0.f);
    float d2acc = 0.f;

    for (int k0 = 0; k0 < D_DIM; k0 += BK) {
        __syncthreads();  // previous chunk's LDS readers done
        // Stage A chunk: targets[srow][k0+skoff .. +15] (coalesced float4)
        {
            const float4* ag = (const float4*)(targets + (size_t)srow * D_DIM + k0 + skoff);
            float4* al = (float4*)(As + srow * AST + skoff);
#pragma unroll
            for (int i = 0; i < 4; ++i) al[i] = ag[i];
        }
        // Stage B chunk: data[col0+srow][k0+skoff .. +15], zero-pad OOB rows
        {
            const int drow = col0 + srow;
            float4* bl = (float4*)(Bs + srow * AST + skoff);
            if (drow < N) {
                const float4* bg = (const float4*)(data + (size_t)drow * D_DIM + k0 + skoff);
#pragma unroll
                for (int i = 0; i < 4; ++i) bl[i] = bg[i];
            } else {
#pragma unroll
                for (int i = 0; i < 4; ++i) bl[i] = fzero;
            }
        }
        __syncthreads();

        // ||d||^2 partial sums from the already-staged chunk (free of HBM traffic)
        if (tid < BN) {
#pragma unroll
            for (int kk = 0; kk < BK; ++kk) {
                float x = Bs[tid * AST + kk];
                d2acc = fmaf(x, x, d2acc);
            }
        }

        // WMMA f32 16x16x4 accumulation. A frag: lanes 0-15 K={kk,kk+1},
        // lanes 16-31 K={kk+2,kk+3}; B frag symmetric (data staged k-major).
        const float* Arow = As + (wave * 16 + lm) * AST;
#pragma unroll
        for (int kk = 0; kk < BK; kk += 4) {
            v2f a = *(const v2f*)(Arow + kk + 2 * half);
#pragma unroll
            for (int ct = 0; ct < 8; ++ct) {
                v2f b = *(const v2f*)(Bs + (ct * 16 + lm) * AST + kk + 2 * half);
                acc[ct] = __builtin_amdgcn_wmma_f32_16x16x4_f32(
                    false, a, false, b, (short)0, acc[ct], false, false);
            }
        }
    }

    __syncthreads();
    if (tid < BN) d2s[tid] = d2acc;
    __syncthreads();

    // Score s = ||d||^2 - 2*(t.d)  (monotone in true distance per row).
    // C layout: VGPR v holds M = v + 8*half, N = lane&15.
#pragma unroll
    for (int ct = 0; ct < 8; ++ct) {
        const int n = ct * 16 + lm;
        const float d2v = d2s[n];
#pragma unroll
        for (int v = 0; v < 8; ++v) {
            const int m = wave * 16 + v + 8 * half;
            Ss[m * SST + n] = d2v - 2.0f * acc[ct][v];
        }
    }
    __syncthreads();

    // Per-row top-K over this block's BN columns (1 thread per row).
    if (tid < NROWS) {
        int valid = N - col0; if (valid > BN) valid = BN;
        float v[TOPK]; int ix[TOPK];
#pragma unroll
        for (int q = 0; q < TOPK; ++q) { v[q] = FLT_MAX; ix[q] = 0; }
        const float* row = Ss + tid * SST;
        for (int j = 0; j < valid; ++j)
            topk_insert(v, ix, row[j], col0 + j);
        const size_t base = ((size_t)tid * nb + blockIdx.x) * TOPK;
#pragma unroll
        for (int q = 0; q < TOPK; ++q) {
            cand_val[base + q] = v[q];
            cand_idx[base + q] = ix[q];
        }
    }
}

// Kernel 2: merge per-block candidate lists -> final 10 indices per row.
__global__ __launch_bounds__(64)
void OTAD_merge(const float* __restrict__ cand_val,
                const int*   __restrict__ cand_idx,
                int* __restrict__ final_idx, int nb)
{
    __shared__ float sv[64 * TOPK];
    __shared__ int   si[64 * TOPK];
    const int row = blockIdx.x;
    const int tid = threadIdx.x;
    const int total = nb * TOPK;
    const float* cv = cand_val + (size_t)row * total;
    const int*   ci = cand_idx + (size_t)row * total;

    float v[TOPK]; int ix[TOPK];
#pragma unroll
    for (int q = 0; q < TOPK; ++q) { v[q] = FLT_MAX; ix[q] = 0; }
    for (int i = tid; i < total; i += 64)
        topk_insert(v, ix, cv[i], ci[i]);
#pragma unroll
    for (int q = 0; q < TOPK; ++q) { sv[tid * TOPK + q] = v[q]; si[tid * TOPK + q] = ix[q]; }
    __syncthreads();
    if (tid == 0) {
        float fv[TOPK]; int fx[TOPK];
#pragma unroll
        for (int q = 0; q < TOPK; ++q) { fv[q] = FLT_MAX; fx[q] = 0; }
        for (int i = 0; i < 64 * TOPK; ++i)
            topk_insert(fv, fx, sv[i], si[i]);
#pragma unroll
        for (int q = 0; q < TOPK; ++q) final_idx[row * TOPK + q] = fx[q];
    }
}

// Kernel 3: gather OTinput/OToutput rows into the concatenated output.
__global__ __launch_bounds__(256)
void OTAD_gather(const float* __restrict__ OTin,   // [N, DIN]
                 const float* __restrict__ OTout,  // [N, DOUT]
                 const int*   __restrict__ final_idx,
                 float* __restrict__ out,
                 int DIN, int DOUT, size_t out2_off)
{
    const int pair = blockIdx.x;            // b*TOPK + k
    const int idx  = final_idx[pair];
    const float4* s1 = (const float4*)(OTin + (size_t)idx * DIN);
    float4* d1 = (float4*)(out + (size_t)pair * DIN);
    const int n1 = DIN >> 2;
    for (int i = threadIdx.x; i < n1; i += blockDim.x) d1[i] = s1[i];
    const float4* s2 = (const float4*)(OTout + (size_t)idx * DOUT);
    float4* d2 = (float4*)(out + out2_off + (size_t)pair * DOUT);
    const int n2 = DOUT >> 2;
    for (int i = threadIdx.x; i < n2; i += blockDim.x) d2[i] = s2[i];
}

extern "C" void kernel_launch(void* const* d_in, const int* in_sizes, int n_in,
                              void* d_out, int out_size, void* d_ws, size_t ws_size,
                              hipStream_t stream)
{
    const float* targets = (const float*)d_in[0];
    const float* data    = (const float*)d_in[1];
    const float* OTin    = (const float*)d_in[2];
    const float* OTout   = (const float*)d_in[3];

    const int D  = D_DIM;                       // 384
    const int B  = in_sizes[0] / D;             // 128
    const int N  = in_sizes[1] / D;             // 50000
    const int DO = (N > 0) ? (in_sizes[3] / N) : 0;  // 2048
    const int nb = (N + BN - 1) / BN;           // 391

    // Workspace layout: [cand_val f32][cand_idx i32][final_idx i32]
    float* cand_val  = (float*)d_ws;
    int*   cand_idx  = (int*)(cand_val + (size_t)B * nb * TOPK);
    int*   final_idx = (int*)(cand_idx + (size_t)B * nb * TOPK);

    const size_t shmem = (size_t)(NROWS * SST + BN) * sizeof(float); // ~65 KB (>= staging 36 KB)
    OTAD_dist_topk<<<nb, 256, shmem, stream>>>(targets, data, cand_val, cand_idx, N, nb);
    OTAD_merge<<<B, 64, 0, stream>>>(cand_val, cand_idx, final_idx, nb);

    const size_t out2_off = (size_t)B * TOPK * D;
    OTAD_gather<<<B * TOPK, 256, 0, stream>>>(OTin, OTout, final_idx,
                                              (float*)d_out, D, DO, out2_off);
}